// GAT_87058987090161
// MI455X (gfx1250) — compile-verified
//
#include <hip/hip_runtime.h>
#include <hip/hip_bf16.h>
#include <math.h>

#define NEG_SLOPE 0.2f

typedef float v2f __attribute__((ext_vector_type(2)));
typedef float v8f __attribute__((ext_vector_type(8)));

// ---------------------------------------------------------------------------
// GEMM:  H[M,F] = X[M,K] @ W[K,F]   via V_WMMA_F32_16X16X4_F32 (exact fp32)
// One wave computes one 16x16 output tile; 8 waves (8 M-tiles) per block share
// the same 16-column W slice (L1-friendly). A/B/C fragment layouts follow the
// CDNA5 ISA 32-bit WMMA VGPR mappings (§7.12.2).
// ---------------------------------------------------------------------------
__global__ void __launch_bounds__(256)
gat_gemm_wmma(const float* __restrict__ X, const float* __restrict__ W,
              float* __restrict__ H, int M, int K, int F) {
  const int lane  = threadIdx.x & 31;
  const int wave  = threadIdx.x >> 5;
  const int tileN = blockIdx.x;                 // F / 16 column tiles
  const int tileM = blockIdx.y * 8 + wave;      // 8 row tiles per block
  if (tileM * 16 >= M) return;                  // whole-wave uniform exit

  const int r    = lane & 15;                   // A row / B,C column in tile
  const int kh   = (lane >> 4) * 2;             // K sub-offset: 0 or 2
  const int mrow = tileM * 16 + r;
  const int ncol = tileN * 16 + r;

  const float* __restrict__ xrow = X + (size_t)mrow * K;
  v8f acc = {};
  for (int k = 0; k < K; k += 4) {
    v2f a, b;
    a.x = xrow[k + kh];                           // A: 16x4, row=lane&15
    a.y = xrow[k + kh + 1];
    b.x = W[(size_t)(k + kh) * F + ncol];         // B: 4x16, col=lane&15
    b.y = W[(size_t)(k + kh + 1) * F + ncol];
    acc = __builtin_amdgcn_wmma_f32_16x16x4_f32(false, a, false, b,
                                                (short)0, acc, false, false);
  }
  const int rblk = (lane >> 4) * 8;               // C/D: VGPR v -> M = v (+8)
#pragma unroll
  for (int v = 0; v < 8; ++v) {
    H[(size_t)(tileM * 16 + rblk + v) * F + ncol] = acc[v];
  }
}

// ---------------------------------------------------------------------------
// alpha_s[n,h] = sum_c h[n,h,c]*a_src[h,c];  alpha_d likewise
// ---------------------------------------------------------------------------
__global__ void gat_node_alpha(const float* __restrict__ h,
                               const float* __restrict__ a_s,
                               const float* __restrict__ a_d,
                               float* __restrict__ out_s,
                               float* __restrict__ out_d,
                               int N, int Hh, int C) {
  int idx = blockIdx.x * blockDim.x + threadIdx.x;
  if (idx >= N * Hh) return;
  int n = idx / Hh, hd = idx % Hh;
  const float* hp  = h + ((size_t)n * Hh + hd) * C;
  const float* asp = a_s + hd * C;
  const float* adp = a_d + hd * C;
  float ss = 0.f, sd = 0.f;
#pragma unroll 4
  for (int c = 0; c < C; ++c) {
    float v = hp[c];
    ss += v * asp[c];
    sd += v * adp[c];
  }
  out_s[idx] = ss;
  out_d[idx] = sd;
}

// ---------------------------------------------------------------------------
// helpers
// ---------------------------------------------------------------------------
__global__ void gat_fill(float* __restrict__ p, long n, float v) {
  long i = (long)blockIdx.x * blockDim.x + threadIdx.x;
  if (i < n) p[i] = v;
}

__device__ __forceinline__ int edge_src(const int* ei, int e, int E) {
  return (e < E) ? ei[e] : (e - E);                 // implicit self-loops
}
__device__ __forceinline__ int edge_dst(const int* ei, int e, int E) {
  return (e < E) ? ei[E + e] : (e - E);
}
__device__ __forceinline__ float leaky(float x) {
  return x > 0.f ? x : NEG_SLOPE * x;
}
// float atomic max via ordered-int trick (init target to -INF):
// lowers to global_atomic_max_i32 / global_atomic_min_u32
__device__ __forceinline__ void atomic_max_f32(float* a, float v) {
  if (v >= 0.f) atomicMax((int*)a, __float_as_int(v));
  else          atomicMin((unsigned int*)a, __float_as_uint(v));
}

// ---------------------------------------------------------------------------
// pass A: segment max of leaky-relu logits over dst
// ---------------------------------------------------------------------------
__global__ void gat_edge_max(const int* __restrict__ ei,
                             const float* __restrict__ a_s,
                             const float* __restrict__ a_d,
                             float* __restrict__ smax, int E, int N, int Hh) {
  long idx = (long)blockIdx.x * blockDim.x + threadIdx.x;
  long tot = (long)(E + N) * Hh;
  if (idx >= tot) return;
  int e = (int)(idx / Hh), hd = (int)(idx % Hh);
  int s = edge_src(ei, e, E), d = edge_dst(ei, e, E);
  float lg = leaky(a_s[s * Hh + hd] + a_d[d * Hh + hd]);
  atomic_max_f32(&smax[d * Hh + hd], lg);
}

// ---------------------------------------------------------------------------
// pass B: segment sum of exp(logit - max)
// ---------------------------------------------------------------------------
__global__ void gat_edge_sum(const int* __restrict__ ei,
                             const float* __restrict__ a_s,
                             const float* __restrict__ a_d,
                             const float* __restrict__ smax,
                             float* __restrict__ ssum, int E, int N, int Hh) {
  long idx = (long)blockIdx.x * blockDim.x + threadIdx.x;
  long tot = (long)(E + N) * Hh;
  if (idx >= tot) return;
  int e = (int)(idx / Hh), hd = (int)(idx % Hh);
  int s = edge_src(ei, e, E), d = edge_dst(ei, e, E);
  float lg = leaky(a_s[s * Hh + hd] + a_d[d * Hh + hd]);
  float m = smax[d * Hh + hd];
  if (m == -INFINITY) m = 0.f;                     // reference isfinite guard
  atomicAdd(&ssum[d * Hh + hd], __expf(lg - m));
}

// ---------------------------------------------------------------------------
// pass C: one wave per edge.  Recompute alpha (4 exps) instead of spilling a
// 13MB edge buffer, then coalesced gather of h[src] and float atomic
// scatter-add into acc[dst].  All traffic is L2-resident on MI455X.
// ---------------------------------------------------------------------------
__global__ void __launch_bounds__(256)
gat_edge_scatter(const int* __restrict__ ei, const float* __restrict__ h,
                 const float* __restrict__ a_s, const float* __restrict__ a_d,
                 const float* __restrict__ smax, const float* __restrict__ ssum,
                 float* __restrict__ acc, int E, int N, int Hh, int C) {
  int gw   = (int)(((long)blockIdx.x * blockDim.x + threadIdx.x) >> 5);
  int lane = threadIdx.x & 31;
  if (gw >= E + N) return;
  int s = edge_src(ei, gw, E), d = edge_dst(ei, gw, E);

  float alpha[4] = {0.f, 0.f, 0.f, 0.f};
  for (int hd = 0; hd < Hh; ++hd) {
    float lg = leaky(a_s[s * Hh + hd] + a_d[d * Hh + hd]);
    float m = smax[d * Hh + hd];
    if (m == -INFINITY) m = 0.f;
    float ex = __expf(lg - m);
    alpha[hd] = ex / (ssum[d * Hh + hd] + 1e-16f);
  }

  const int F   = Hh * C;
  const int csh = (C == 64) ? 6 : 7;               // f / C as shift
  const float* __restrict__ hs = h + (size_t)s * F;
  float* __restrict__ op = acc + (size_t)d * F;
  for (int f = lane; f < F; f += 32) {
    atomicAdd(&op[f], hs[f] * alpha[f >> csh]);    // global_atomic_add_f32
  }
}

// ---------------------------------------------------------------------------
// out += bias; optional relu (in place)
// ---------------------------------------------------------------------------
__global__ void gat_bias_act(float* __restrict__ p, const float* __restrict__ b,
                             long total, int F, int do_relu) {
  long i = (long)blockIdx.x * blockDim.x + threadIdx.x;
  if (i >= total) return;
  float v = p[i] + b[(int)(i % F)];
  p[i] = do_relu ? fmaxf(v, 0.f) : v;
}

// ---------------------------------------------------------------------------
static void run_layer(hipStream_t stream, const float* xin, int K,
                      const float* W, const float* a_s, const float* a_d,
                      const float* b, int Hh, int C, int do_relu,
                      const int* ei, int N, int E,
                      float* hfeat, float* acc,
                      float* aS, float* aD, float* smax, float* ssum) {
  const int F = Hh * C;
  const int mtiles = (N + 15) / 16;
  dim3 ggrid(F / 16, (mtiles + 7) / 8);
  gat_gemm_wmma<<<ggrid, 256, 0, stream>>>(xin, W, hfeat, N, K, F);

  const int nh = N * Hh;
  gat_node_alpha<<<(nh + 255) / 256, 256, 0, stream>>>(hfeat, a_s, a_d,
                                                       aS, aD, N, Hh, C);
  gat_fill<<<(nh + 255) / 256, 256, 0, stream>>>(smax, nh, -INFINITY);
  gat_fill<<<(nh + 255) / 256, 256, 0, stream>>>(ssum, nh, 0.f);

  const long accn = (long)N * F;
  gat_fill<<<(int)((accn + 255) / 256), 256, 0, stream>>>(acc, accn, 0.f);

  const long eh = (long)(E + N) * Hh;
  gat_edge_max<<<(int)((eh + 255) / 256), 256, 0, stream>>>(ei, aS, aD, smax,
                                                            E, N, Hh);
  gat_edge_sum<<<(int)((eh + 255) / 256), 256, 0, stream>>>(ei, aS, aD, smax,
                                                            ssum, E, N, Hh);
  const long lanes = (long)(E + N) * 32;
  gat_edge_scatter<<<(int)((lanes + 255) / 256), 256, 0, stream>>>(
      ei, hfeat, aS, aD, smax, ssum, acc, E, N, Hh, C);

  gat_bias_act<<<(int)((accn + 255) / 256), 256, 0, stream>>>(acc, b, accn, F,
                                                              do_relu);
}

extern "C" void kernel_launch(void* const* d_in, const int* in_sizes, int n_in,
                              void* d_out, int out_size, void* d_ws,
                              size_t ws_size, hipStream_t stream) {
  const float* x   = (const float*)d_in[0];
  const int*   ei  = (const int*)d_in[1];     // [2, E] int32
  const float* W0  = (const float*)d_in[2];
  const float* as0 = (const float*)d_in[3];
  const float* ad0 = (const float*)d_in[4];
  const float* b0  = (const float*)d_in[5];
  const float* W1  = (const float*)d_in[6];
  const float* as1 = (const float*)d_in[7];
  const float* ad1 = (const float*)d_in[8];
  const float* b1  = (const float*)d_in[9];
  const float* W2  = (const float*)d_in[10];
  const float* as2 = (const float*)d_in[11];
  const float* ad2 = (const float*)d_in[12];
  const float* b2  = (const float*)d_in[13];

  const int N = in_sizes[0] / 128;
  const int E = in_sizes[1] / 2;

  // workspace: two N*256 feature buffers + 4 small N*4 vectors (~106 MB)
  float* bufX = (float*)d_ws;                    // layer input / scatter acc
  float* bufF = bufX + (size_t)N * 256;          // GEMM output (h features)
  float* aS   = bufF + (size_t)N * 256;
  float* aD   = aS + (size_t)N * 4;
  float* smax = aD + (size_t)N * 4;
  float* ssum = smax + (size_t)N * 4;

  // layer input becomes dead after the GEMM reads it, so the scatter
  // accumulator aliases it (stream order serializes GEMM -> fill -> scatter).
  run_layer(stream, x,    128, W0, as0, ad0, b0, 4, 64, 1, ei, N, E,
            bufF, bufX, aS, aD, smax, ssum);
  run_layer(stream, bufX, 256, W1, as1, ad1, b1, 4, 64, 1, ei, N, E,
            bufF, bufX, aS, aD, smax, ssum);
  // heads=1, mean over heads == identity; scatter straight into d_out
  run_layer(stream, bufX, 256, W2, as2, ad2, b2, 1, 128, 0, ei, N, E,
            bufF, (float*)d_out, aS, aD, smax, ssum);
}